// ChamferDistance_56831007261041
// MI455X (gfx1250) — compile-verified
//
#include <hip/hip_runtime.h>
#include <math.h>

#define EPS 1e-8f

typedef __attribute__((ext_vector_type(2))) float v2f;
typedef __attribute__((ext_vector_type(8))) float v8f;

// ---------------------------------------------------------------------------
// Kernel 1: pack both clouds into WMMA A-side and B-side fragment layouts.
// K=4 trick: D = A x B accumulates  -2*dot(p,q) + ||q||^2  directly:
//   A row  = (-2px, -2py | -2pz, 1)      (half 0 | half 1)
//   B col  = (  qx,   qy |   qz, ||q||^2)
// Layout: pk[(batch*2 + half)*n + idx] = float2 fragment for that half.
// ---------------------------------------------------------------------------
__global__ void chamfer_pack(const float* __restrict__ S,   // (B*N,3)
                             const float* __restrict__ T,   // (B*M,3)
                             float2* __restrict__ SA, float2* __restrict__ SB, // (B,2,N)
                             float2* __restrict__ TA, float2* __restrict__ TB, // (B,2,M)
                             float* __restrict__ ns,        // (B*N)
                             float* __restrict__ nt,        // (B*M)
                             int B, int N, int M) {
    int i = blockIdx.x * blockDim.x + threadIdx.x;
    int stride = gridDim.x * blockDim.x;
    int BN = B * N, BM = B * M;
    for (int k = i; k < BN; k += stride) {
        int b = k / N, idx = k - b * N;
        float x = S[k * 3 + 0], y = S[k * 3 + 1], z = S[k * 3 + 2];
        float nrm = fmaf(x, x, fmaf(y, y, z * z));
        ns[k] = nrm;
        size_t h0 = ((size_t)b * 2 + 0) * N + idx;
        size_t h1 = ((size_t)b * 2 + 1) * N + idx;
        SA[h0] = make_float2(-2.0f * x, -2.0f * y);
        SA[h1] = make_float2(-2.0f * z, 1.0f);
        SB[h0] = make_float2(x, y);
        SB[h1] = make_float2(z, nrm);
    }
    for (int k = i; k < BM; k += stride) {
        int b = k / M, idx = k - b * M;
        float x = T[k * 3 + 0], y = T[k * 3 + 1], z = T[k * 3 + 2];
        float nrm = fmaf(x, x, fmaf(y, y, z * z));
        nt[k] = nrm;
        size_t h0 = ((size_t)b * 2 + 0) * M + idx;
        size_t h1 = ((size_t)b * 2 + 1) * M + idx;
        TA[h0] = make_float2(-2.0f * x, -2.0f * y);
        TA[h1] = make_float2(-2.0f * z, 1.0f);
        TB[h0] = make_float2(x, y);
        TB[h1] = make_float2(z, nrm);
    }
}

// ---------------------------------------------------------------------------
// Kernel 2: one wave owns one 16-row tile (A side) and sweeps all columns
// (B side). Inner loop: 1x global_load_b64 + 1x V_WMMA_F32_16X16X4_F32 +
// 8 mins. D already holds ||q||^2 - 2 p.q, so no FMA epilogue at all.
// Deferred monotonic + ||p||^2 and clamp after the loop, then shfl-xor
// column-min reduction. Each row tile has one owner -> plain stores.
// ---------------------------------------------------------------------------
__global__ __launch_bounds__(256)
void chamfer_min_rows(const float2* __restrict__ Apk,   // (B,2,nP) A-form rows
                      const float2* __restrict__ Bpk,   // (B,2,nQ) B-form cols
                      const float* __restrict__ nrmP,   // (B*nP)
                      float* __restrict__ minOut,       // (B*nP)
                      int nP, int nQ, int nWaves) {
    const int waveInBlock = threadIdx.x >> 5;
    const int lane        = threadIdx.x & 31;
    const int gw          = blockIdx.x * 8 + waveInBlock;
    if (gw >= nWaves) return;                 // uniform per wave: EXEC stays all-1s

    const int tilesPerBatch = nP >> 4;
    const int batch = gw / tilesPerBatch;
    const int tile  = gw - batch * tilesPerBatch;

    const int half = lane >> 4;               // 0: K=0..1 ; 1: K=2..3
    const int lrow = lane & 15;
    const int i0   = tile * 16;

    const float2* prow = Apk + ((size_t)batch * 2 + half) * nP;
    const float2* qcol = Bpk + ((size_t)batch * 2 + half) * nQ + lrow;
    const float*  nPb  = nrmP + (size_t)batch * nP + i0 + 8 * half;

    float2 ap = prow[i0 + lrow];              // one b64 load
    v2f a;
    a.x = ap.x;
    a.y = ap.y;

    // Squared norms of the 8 rows this lane's D fragment covers.
    float nsv[8];
#pragma unroll
    for (int v = 0; v < 8; ++v) nsv[v] = nPb[v];

    float rmin[8];
#pragma unroll
    for (int v = 0; v < 8; ++v) rmin[v] = 3.402823466e38f;

    for (int j0 = 0; j0 < nQ; j0 += 16) {
        float2 t = qcol[j0];                  // one b64 load
        v2f b;
        b.x = t.x;
        b.y = t.y;

        v8f c = {};
        // d[v] = ||q_col||^2 - 2 * dot(p_row, q_col)   (norm folded into K=3)
        v8f d = __builtin_amdgcn_wmma_f32_16x16x4_f32(
            false, a, false, b, (short)0, c, false, false);

#pragma unroll
        for (int v = 0; v < 8; ++v)
            rmin[v] = fminf(rmin[v], d[v]);
    }

    // Deferred monotonic terms: + ||p||^2, clamp tiny negatives.
#pragma unroll
    for (int v = 0; v < 8; ++v) rmin[v] = fmaxf(rmin[v] + nsv[v], 0.0f);

    // Row-min across the 16 columns = reduce across the 16 lanes of each half.
#pragma unroll
    for (int m = 8; m >= 1; m >>= 1) {
#pragma unroll
        for (int v = 0; v < 8; ++v)
            rmin[v] = fminf(rmin[v], __shfl_xor(rmin[v], m, 32));
    }

    if (lrow == 0) {                          // lanes 0 and 16 write 8 rows each
        float* mO = minOut + (size_t)batch * nP + i0 + 8 * half;
#pragma unroll
        for (int v = 0; v < 8; ++v) mO[v] = rmin[v];
    }
}

// ---------------------------------------------------------------------------
// Kernel 3: single-workgroup deterministic reduction ->
//   out = mean(sqrt(minS+eps) * w) + mean(sqrt(minT+eps))
// ---------------------------------------------------------------------------
__global__ __launch_bounds__(1024)
void chamfer_reduce(const float* __restrict__ minS,
                    const float* __restrict__ minT,
                    const float* __restrict__ w,
                    float* __restrict__ out,
                    int BN, int BM, float invBN, float invBM) {
    __shared__ float sh[1024];
    const int tid = threadIdx.x;
    float acc = 0.0f;
    for (int i = tid; i < BN; i += 1024)
        acc += sqrtf(minS[i] + EPS) * w[i] * invBN;
    for (int i = tid; i < BM; i += 1024)
        acc += sqrtf(minT[i] + EPS) * invBM;
    sh[tid] = acc;
    __syncthreads();
#pragma unroll
    for (int s = 512; s > 0; s >>= 1) {
        if (tid < s) sh[tid] += sh[tid + s];
        __syncthreads();
    }
    if (tid == 0) out[0] = sh[0];
}

// ---------------------------------------------------------------------------
extern "C" void kernel_launch(void* const* d_in, const int* in_sizes, int n_in,
                              void* d_out, int out_size, void* d_ws, size_t ws_size,
                              hipStream_t stream) {
    const float* src = (const float*)d_in[0];   // (B,N,3)
    const float* tgt = (const float*)d_in[1];   // (B,M,3)
    const float* wgt = (const float*)d_in[2];   // (B,N)

    const int B = 4;
    const int N = in_sizes[0] / (B * 3);
    const int M = in_sizes[1] / (B * 3);
    const int BN = B * N;
    const int BM = B * M;

    // Workspace layout (float2 arrays first; 8B alignment from base).
    float2* SA = (float2*)d_ws;                  // B*2*N
    float2* SB = SA + (size_t)2 * BN;            // B*2*N
    float2* TA = SB + (size_t)2 * BN;            // B*2*M
    float2* TB = TA + (size_t)2 * BM;            // B*2*M
    float*  ns   = (float*)(TB + (size_t)2 * BM);// B*N
    float*  nt   = ns + BN;                      // B*M
    float*  minS = nt + BM;                      // B*N
    float*  minT = minS + BN;                    // B*M

    // 1) pack + norms
    {
        int total = (BN > BM) ? BN : BM;
        int blocks = (total + 255) / 256;
        chamfer_pack<<<blocks, 256, 0, stream>>>(src, tgt, SA, SB, TA, TB,
                                                 ns, nt, B, N, M);
    }

    // 2) WMMA min passes (row-direction mins; swap roles for the other direction)
    {
        int nWavesF = B * (N / 16);
        int blocksF = (nWavesF + 7) / 8;
        chamfer_min_rows<<<blocksF, 256, 0, stream>>>(SA, TB, ns, minS, N, M, nWavesF);

        int nWavesB = B * (M / 16);
        int blocksB = (nWavesB + 7) / 8;
        chamfer_min_rows<<<blocksB, 256, 0, stream>>>(TA, SB, nt, minT, M, N, nWavesB);
    }

    // 3) deterministic final reduction
    chamfer_reduce<<<1, 1024, 0, stream>>>(minS, minT, wgt, (float*)d_out,
                                           BN, BM, 1.0f / BN, 1.0f / BM);
}